// OutputPPBlockSUF_32384053412125
// MI455X (gfx1250) — compile-verified
//
#include <hip/hip_runtime.h>
#include <math.h>

// ---------------------------------------------------------------------------
// OutputPPBlock for MI455X (gfx1250, wave32, WMMA)
//
// Phase 1 (bandwidth-bound): msg = (rbf @ W_rbf) * x, scatter-added into a
//   [N,256] f32 node accumulator held in d_ws (30 MB -> L2 resident).
//   512 MB of x traffic @ 23.3 TB/s dominates; atomics land in L2.
// Phase 2 (matrix-bound): per branch, 4 chained 256x256 GEMMs over 30k nodes
//   (31.5 GFLOP f32) using V_WMMA_F32_16X16X4_F32, activations staged in LDS.
//   SiLU uses v_exp_f32 + v_rcp_f32 (no IEEE divide sequence).
// ---------------------------------------------------------------------------

typedef float v2f __attribute__((ext_vector_type(2)));
typedef float v8f __attribute__((ext_vector_type(8)));

#define H_DIM   256
#define OE_DIM  256
#define R_DIM   6
#define A_BR    2
#define L_LAY   3
#define TM      32            // nodes per workgroup tile
#define YPITCH  260           // LDS row pitch (floats): 256 + 4 pad -> bank-conflict-free

static __device__ __forceinline__ v8f wmma_f32_4(v2f a, v2f b, v8f c) {
    // D = A(16x4 f32) * B(4x16 f32) + C(16x16 f32)
    return __builtin_amdgcn_wmma_f32_16x16x4_f32(
        /*neg_a=*/false, a, /*neg_b=*/false, b,
        /*c_mod=*/(short)0, c, /*reuse_a=*/false, /*reuse_b=*/false);
}

static __device__ __forceinline__ float silu_fast(float v) {
    // x * sigmoid(x) with v_exp_f32 + v_rcp_f32 (avoids div-scale/fixup chain)
    return v * __builtin_amdgcn_rcpf(1.f + __expf(-v));
}

// ---------------------------------------------------------------------------
// Phase 1: one block per edge; thread = one H column.
// ---------------------------------------------------------------------------
__global__ __launch_bounds__(256) void edge_scatter_kernel(
        const float* __restrict__ x,      // [E, H]
        const float* __restrict__ rbf,    // [E, R]
        const int*   __restrict__ idx,    // [E]
        const float* __restrict__ W_rbf,  // [R, H]
        float*       __restrict__ hbuf,   // [N, H] accumulator (pre-zeroed)
        int E) {
    const int e = blockIdx.x;
    if (e >= E) return;
    const int hc = threadIdx.x;

    float proj = 0.f;
#pragma unroll
    for (int r = 0; r < R_DIM; ++r)
        proj = fmaf(rbf[(size_t)e * R_DIM + r], W_rbf[r * H_DIM + hc], proj);

    const float msg  = proj * x[(size_t)e * H_DIM + hc];
    const int   node = idx[e];
    // Lowers to global_atomic_add_f32 (no-return) -> STOREcnt path, L2 RMW.
    unsafeAtomicAdd(&hbuf[(size_t)node * H_DIM + hc], msg);
}

// ---------------------------------------------------------------------------
// Phase 2: 32-node tile MLP. 8 waves; wave w owns N-tiles {2w, 2w+1} and both
// 16-row M-subtiles (B fragments shared across the two M-subtiles).
// ---------------------------------------------------------------------------
__global__ __launch_bounds__(256) void mlp_wmma_kernel(
        const float* __restrict__ hbuf,      // [N, H]
        const float* __restrict__ W_up,      // [A, H, OE]
        const float* __restrict__ W_hidden,  // [A, L, OE, OE]
        const float* __restrict__ b_hidden,  // [A, L, OE]
        const float* __restrict__ W_out,     // [A, OE, 1]
        float*       __restrict__ out,       // [A, N]
        int N) {
    __shared__ __align__(16) float ylds[TM][YPITCH];

    const int a    = blockIdx.y;
    const int n0   = blockIdx.x * TM;
    const int tid  = threadIdx.x;
    const int wave = tid >> 5;
    const int lane = tid & 31;
    const int half = lane >> 4;   // 0: lanes 0-15, 1: lanes 16-31
    const int tn   = lane & 15;   // fragment row (A) / col (B,C)
    const int nt0  = wave * 2;    // first of this wave's two N-tiles

    // ---- stage h tile into LDS (coalesced) ----
    for (int t = tid; t < TM * H_DIM; t += 256) {
        const int row = t >> 8;
        const int col = t & 255;
        const int g   = n0 + row;
        ylds[row][col] = (g < N) ? hbuf[(size_t)g * H_DIM + col] : 0.f;
    }
    __syncthreads();

    // ---- 4 GEMM layers: up (no bias/act) + 3 hidden (bias + SiLU) ----
#pragma unroll 1
    for (int layer = 0; layer < 1 + L_LAY; ++layer) {
        const bool   act = (layer > 0);
        const float* Wl  = (layer == 0)
            ? (W_up + (size_t)a * H_DIM * OE_DIM)
            : (W_hidden + ((size_t)(a * L_LAY + (layer - 1))) * OE_DIM * OE_DIM);

        float bias0 = 0.f, bias1 = 0.f;
        if (act) {
            const float* bl = b_hidden + (size_t)(a * L_LAY + (layer - 1)) * OE_DIM;
            bias0 = bl[(nt0 + 0) * 16 + tn];
            bias1 = bl[(nt0 + 1) * 16 + tn];
        }

        v8f c00, c01, c10, c11;  // [mi][nti]
#pragma unroll
        for (int j = 0; j < 8; ++j) {
            c00[j] = bias0; c10[j] = bias0;
            c01[j] = bias1; c11[j] = bias1;
        }

#pragma unroll 4
        for (int k = 0; k < H_DIM; k += 4) {
            // A fragments from LDS: lane row M = tn, k-pair selected by half.
            const int kc = k + 2 * half;
            const v2f a0 = *(const v2f*)&ylds[tn][kc];        // M-subtile 0
            const v2f a1 = *(const v2f*)&ylds[16 + tn][kc];   // M-subtile 1

            // B fragments from global (L2-hot weights): B[kc..kc+1][n]
            const float* bp0 = Wl + (size_t)kc * OE_DIM + (nt0 + 0) * 16 + tn;
            const float* bp1 = Wl + (size_t)kc * OE_DIM + (nt0 + 1) * 16 + tn;
            v2f b0, b1;
            b0.x = bp0[0]; b0.y = bp0[OE_DIM];
            b1.x = bp1[0]; b1.y = bp1[OE_DIM];

            c00 = wmma_f32_4(a0, b0, c00);
            c10 = wmma_f32_4(a1, b0, c10);
            c01 = wmma_f32_4(a0, b1, c01);
            c11 = wmma_f32_4(a1, b1, c11);
        }

        __syncthreads();  // all waves done reading ylds for this layer

        // Writeback (+ fused SiLU): C elem j -> row = mi*16 + 8*half + j, col = nt*16+tn
#pragma unroll
        for (int j = 0; j < 8; ++j) {
            const int r0 = 8 * half + j;        // M-subtile 0 row
            const int r1 = 16 + 8 * half + j;   // M-subtile 1 row
            float v00 = c00[j], v01 = c01[j], v10 = c10[j], v11 = c11[j];
            if (act) {
                v00 = silu_fast(v00);
                v01 = silu_fast(v01);
                v10 = silu_fast(v10);
                v11 = silu_fast(v11);
            }
            ylds[r0][(nt0 + 0) * 16 + tn] = v00;
            ylds[r0][(nt0 + 1) * 16 + tn] = v01;
            ylds[r1][(nt0 + 0) * 16 + tn] = v10;
            ylds[r1][(nt0 + 1) * 16 + tn] = v11;
        }
        __syncthreads();
    }

    // ---- final 256 -> 1 projection: 8 lanes per node, shuffle reduce ----
    const int node = tid >> 3;
    const int kb   = (tid & 7) * 32;
    const float* wo = W_out + (size_t)a * OE_DIM;
    float s = 0.f;
#pragma unroll 8
    for (int j = 0; j < 32; ++j)
        s = fmaf(ylds[node][kb + j], wo[kb + j], s);
    s += __shfl_down(s, 4, 8);
    s += __shfl_down(s, 2, 8);
    s += __shfl_down(s, 1, 8);
    if ((tid & 7) == 0 && (n0 + node) < N)
        out[(size_t)a * N + n0 + node] = s;
}

// ---------------------------------------------------------------------------
extern "C" void kernel_launch(void* const* d_in, const int* in_sizes, int n_in,
                              void* d_out, int out_size, void* d_ws, size_t ws_size,
                              hipStream_t stream) {
    const float* x        = (const float*)d_in[0];  // [E, H]
    const float* rbf      = (const float*)d_in[1];  // [E, R]
    const int*   idx      = (const int*)  d_in[2];  // [E]
    // d_in[3] = num_nodes device scalar (N recovered from out_size instead)
    const float* W_rbf    = (const float*)d_in[4];  // [R, H]
    const float* W_up     = (const float*)d_in[5];  // [A, H, OE]
    const float* W_hidden = (const float*)d_in[6];  // [A, L, OE, OE]
    const float* b_hidden = (const float*)d_in[7];  // [A, L, OE]
    const float* W_out    = (const float*)d_in[8];  // [A, OE, 1]
    float*       out      = (float*)d_out;

    const int E = in_sizes[0] / H_DIM;
    const int N = out_size / A_BR;          // OUT == 1, two branches

    float* hbuf = (float*)d_ws;             // [N, H] accumulator
    hipMemsetAsync(hbuf, 0, (size_t)N * H_DIM * sizeof(float), stream);

    edge_scatter_kernel<<<dim3(E), dim3(256), 0, stream>>>(x, rbf, idx, W_rbf, hbuf, E);

    const int nblocks = (N + TM - 1) / TM;
    mlp_wmma_kernel<<<dim3(nblocks, A_BR), dim3(256), 0, stream>>>(
        hbuf, W_up, W_hidden, b_hidden, W_out, out, N);
}